// Upsample1d_16741782519815
// MI455X (gfx1250) — compile-verified
//
#include <hip/hip_runtime.h>
#include <hip/hip_bf16.h>

// Upsample1d (linear kernel, reflect pad, stride-2 transposed depthwise FIR)
// B=8, C=512, L=8192  ->  out [B, C, 2L], float32.
//
// Math (per row, taps w = [wA, wB, wB, wA] = [0.25, 0.75, 0.75, 0.25]):
//   out[2t]   = wA*x[t-1] + wB*x[t]     (x[-1] := x[1])
//   out[2t+1] = wB*x[t]   + wA*x[t+1]   (x[L]  := x[L-2])
//
// Bandwidth bound: 384 MB @ 23.3 TB/s ~= 16.5 us floor. Strategy: async
// b128 loads into LDS (ASYNCcnt path), LDS halo for the stencil, 128-bit
// non-temporal stores (write-once stream larger than the 192 MB L2).

#define L_IN   8192
#define L_OUT  (2 * L_IN)
#define TILE   2048          // input floats per block
#define TPB    256           // 8 waves of 32

typedef float v4f __attribute__((ext_vector_type(4)));

__global__ __launch_bounds__(TPB)
void upsample1d_async_kernel(const float* __restrict__ in,
                             const float* __restrict__ kern,
                             float* __restrict__ out) {
    // s[0..TILE-1] = x[t0 .. t0+TILE-1]; s[TILE] = left halo; s[TILE+1] = right halo
    __shared__ __align__(16) float s[TILE + 2];

    const int tile = blockIdx.x;              // 0..3
    const int row  = blockIdx.y;              // 0..4095
    const int t0   = tile * TILE;
    const int tid  = threadIdx.x;
    const int j    = tid * 8;                 // this thread's 8-float slot in s[]

    const float* in_row  = in  + (size_t)row * L_IN;
    float*       out_row = out + (size_t)row * L_OUT;

    // Symmetric 4-tap kernel: taps are [wA, wB, wB, wA]; uniform scalar loads.
    const float wA = kern[0];                 // 0.25
    const float wB = kern[1];                 // 0.75

    // ---- async main-tile load: 2x b128 per lane, global -> LDS ----
    {
        unsigned lds0 = (unsigned)(unsigned long long)(&s[j]);   // low 32 bits of the
        unsigned lds1 = lds0 + 16u;                              // generic LDS addr = LDS offset
        const float* g0 = in_row + t0 + j;
        const float* g1 = g0 + 4;
        asm volatile("global_load_async_to_lds_b128 %0, %1, off"
                     :: "v"(lds0), "v"(g0) : "memory");
        asm volatile("global_load_async_to_lds_b128 %0, %1, off"
                     :: "v"(lds1), "v"(g1) : "memory");
    }

    // ---- halo elements (reflect at row boundaries) ----
    if (tid == 0) {
        const int gl = (t0 == 0) ? 1 : (t0 - 1);
        s[TILE] = in_row[gl];
    } else if (tid == 1) {
        const int gr = (t0 + TILE >= L_IN) ? (L_IN - 2) : (t0 + TILE);
        s[TILE + 1] = in_row[gr];
    }

    // Drain this wave's async transfers, then make all LDS writes visible block-wide.
    asm volatile("s_wait_asynccnt 0" ::: "memory");
    __syncthreads();

    // ---- compute 16 outputs from s[j-1 .. j+8] ----
    float x[10];
    x[0] = (j == 0) ? s[TILE] : s[j - 1];
#pragma unroll
    for (int k = 0; k < 8; ++k) x[k + 1] = s[j + k];
    x[9] = (j + 8 == TILE) ? s[TILE + 1] : s[j + 8];

    float e[16];
#pragma unroll
    for (int k = 0; k < 8; ++k) {
        const float a = x[k];                  // x[t-1]
        const float b = x[k + 1];              // x[t]
        const float c = x[k + 2];              // x[t+1]
        e[2 * k]     = fmaf(wA, a, wB * b);    // out[2t]
        e[2 * k + 1] = fmaf(wA, c, wB * b);    // out[2t+1]
    }

    // ---- 4x 16B non-temporal stores, fully coalesced ----
    v4f* op = (v4f*)(out_row + 2 * (t0 + j));
#pragma unroll
    for (int q = 0; q < 4; ++q) {
        v4f v = { e[4 * q + 0], e[4 * q + 1], e[4 * q + 2], e[4 * q + 3] };
        __builtin_nontemporal_store(v, op + q);
    }
}

extern "C" void kernel_launch(void* const* d_in, const int* in_sizes, int n_in,
                              void* d_out, int out_size, void* d_ws, size_t ws_size,
                              hipStream_t stream) {
    const float* x    = (const float*)d_in[0];   // [B, C, L] f32
    const float* kern = (const float*)d_in[1];   // [4] f32
    float*       out  = (float*)d_out;           // [B, C, 2L] f32

    const int rows = in_sizes[0] / L_IN;         // B*C = 4096
    dim3 grid(L_IN / TILE, rows);                // (4, 4096)
    upsample1d_async_kernel<<<grid, TPB, 0, stream>>>(x, kern, out);
}